// GraphRNNEncoder_27659589386698
// MI455X (gfx1250) — compile-verified
//
#include <hip/hip_runtime.h>
#include <hip/hip_bf16.h>

#define N_NODES 512
#define F_FEAT  16
#define B_SZ    8
#define FCOUT   128
#define FCIN    (N_NODES * F_FEAT)

typedef float v2f __attribute__((ext_vector_type(2)));
typedef float v4f __attribute__((ext_vector_type(4)));
typedef float v8f __attribute__((ext_vector_type(8)));

// ---- helpers -------------------------------------------------------------

// Load a 16x16 row-major fp32 matrix into WMMA A-operand layout (K chunks of 4).
// A-layout (f32 16x4): lanes 0-15 row=lane, VGPR0=K0,VGPR1=K1; lanes 16-31: K2,K3.
static __device__ __forceinline__ void load_w16(const float* __restrict__ wp,
                                                int row, int half, v2f a[4]) {
#pragma unroll
  for (int c = 0; c < 4; ++c)
    a[c] = *(const v2f*)(wp + row * F_FEAT + c * 4 + half * 2);
}

// Same, but non-temporal (TH_NT): used for the 256MB one-shot weight streams
// so they do not evict the L2-resident `base` intermediate.
static __device__ __forceinline__ void load_w16_nt(const float* __restrict__ wp,
                                                   int row, int half, v2f a[4]) {
#pragma unroll
  for (int c = 0; c < 4; ++c)
    a[c] = __builtin_nontemporal_load((const v2f*)(wp + row * F_FEAT + c * 4 + half * 2));
}

// D = A(16x16) * B(16x16) + C via 4 chained fp32 WMMAs (K=4 each).
static __device__ __forceinline__ v8f wmma_k16(const v2f a[4], const v2f b[4], v8f acc) {
#pragma unroll
  for (int c = 0; c < 4; ++c)
    acc = __builtin_amdgcn_wmma_f32_16x16x4_f32(false, a[c], false, b[c],
                                                (short)0, acc, false, false);
  return acc;
}

// ---- Phase A: base[b,i,j,f] = sum_g w_nf[i,j,f,g] * x[b,i,g] -------------
// One wave per edge (i,j). Streams w_nf contiguously (256MB, non-temporal).
__global__ void __launch_bounds__(256) phaseA(const float* __restrict__ x,
                                              const float* __restrict__ w_nf,
                                              float* __restrict__ base) {
  const int lane = threadIdx.x & 31;
  const int wave = threadIdx.x >> 5;
  const int col  = lane & 15;   // B column = batch b
  const int half = lane >> 4;
  const int i = blockIdx.x;
  const int j = blockIdx.y * 8 + wave;

  // B operand: X_i[g, b] (columns >= B_SZ zero)
  v2f bx[4];
#pragma unroll
  for (int c = 0; c < 4; ++c) {
    const int g = c * 4 + half * 2;
    v2f v = {0.f, 0.f};
    if (col < B_SZ)
      v = *(const v2f*)(x + (size_t)col * N_NODES * F_FEAT + (size_t)i * F_FEAT + g);
    bx[c] = v;
  }

  v2f aw[4];
  load_w16_nt(w_nf + ((size_t)i * N_NODES + j) * (F_FEAT * F_FEAT), col, half, aw);

  v8f acc = {};
  acc = wmma_k16(aw, bx, acc);

  if (col < B_SZ) {
    float* bp = base + (((size_t)col * N_NODES + i) * N_NODES + j) * F_FEAT + half * 8;
    v4f d0 = {acc[0], acc[1], acc[2], acc[3]};
    v4f d1 = {acc[4], acc[5], acc[6], acc[7]};
    *(v4f*)(bp)     = d0;
    *(v4f*)(bp + 4) = d1;
  }
}

// ---- Phase B: incoming[b,i,f] = sum_k A[b,i,k]*A[b,k,i]*relu(base[b,k,i,f])
__global__ void __launch_bounds__(256) phaseB(const float* __restrict__ Aadj,
                                              const float* __restrict__ base,
                                              float* __restrict__ incoming) {
  const int i  = blockIdx.x;
  const int b  = blockIdx.y;
  const int f  = threadIdx.x & 15;
  const int kk = threadIdx.x >> 4;   // 0..15

  const float* Ab = Aadj + (size_t)b * N_NODES * N_NODES;
  const float* Bb = base + (size_t)b * N_NODES * N_NODES * F_FEAT;

  float acc = 0.f;
  for (int k = kk; k < N_NODES; k += 16) {
    const float w = Ab[(size_t)i * N_NODES + k] * Ab[(size_t)k * N_NODES + i];
    if (w != 0.f) {
      const float v = Bb[((size_t)k * N_NODES + i) * F_FEAT + f];
      acc += w * fmaxf(v, 0.f);
    }
  }
  __shared__ float red[256];
  red[threadIdx.x] = acc;
  __syncthreads();
#pragma unroll
  for (int s = 128; s >= 16; s >>= 1) {
    if (threadIdx.x < s) red[threadIdx.x] += red[threadIdx.x + s];
    __syncthreads();
  }
  if (threadIdx.x < 16)
    incoming[((size_t)b * N_NODES + i) * F_FEAT + f] = red[threadIdx.x];
}

// ---- Phase C: per edge pre = base + W_nm @ agg; msg_sum[b,i,f] = sum_j A*relu(pre)
// One block per node i; wave w handles j = w, w+8, ... (deterministic order).
__global__ void __launch_bounds__(256) phaseC(const float* __restrict__ Aadj,
                                              const float* __restrict__ w_nm,
                                              const float* __restrict__ base,
                                              const float* __restrict__ incoming,
                                              float* __restrict__ msg_sum) {
  const int lane = threadIdx.x & 31;
  const int wave = threadIdx.x >> 5;
  const int col  = lane & 15;
  const int half = lane >> 4;
  const int i = blockIdx.x;

  // incoming[b,i,g] in B-layout, loaded once per wave
  v2f inc[4];
#pragma unroll
  for (int c = 0; c < 4; ++c) {
    const int g = c * 4 + half * 2;
    v2f v = {0.f, 0.f};
    if (col < B_SZ)
      v = *(const v2f*)(incoming + ((size_t)col * N_NODES + i) * F_FEAT + g);
    inc[c] = v;
  }

  float msum[8];
#pragma unroll
  for (int v = 0; v < 8; ++v) msum[v] = 0.f;

  for (int j = wave; j < N_NODES; j += 8) {
    // prefetch next W_nm tile for this wave
    const int jn = (j + 8 < N_NODES) ? j + 8 : j;
    __builtin_prefetch(w_nm + ((size_t)i * N_NODES + jn) * (F_FEAT * F_FEAT), 0, 1);

    float aij = 0.f, aji = 0.f;
    if (col < B_SZ) {
      const float* Ab = Aadj + (size_t)col * N_NODES * N_NODES;
      aij = Ab[(size_t)i * N_NODES + j];
      aji = Ab[(size_t)j * N_NODES + i];
    }

    // agg[g,b] = incoming[b,i,g] - A[b,i,j] * (A[b,j,i]*relu(base[b,j,i,g]))
    v2f bagg[4];
#pragma unroll
    for (int c = 0; c < 4; ++c) {
      const int g = c * 4 + half * 2;
      v2f bt = {0.f, 0.f};
      if (col < B_SZ)
        bt = *(const v2f*)(base + (((size_t)col * N_NODES + j) * N_NODES + i) * F_FEAT + g);
      v2f agg;
      agg.x = inc[c].x - aij * (aji * fmaxf(bt.x, 0.f));
      agg.y = inc[c].y - aij * (aji * fmaxf(bt.y, 0.f));
      bagg[c] = agg;
    }

    v2f aw[4];
    load_w16_nt(w_nm + ((size_t)i * N_NODES + j) * (F_FEAT * F_FEAT), col, half, aw);

    // C operand = base[b,i,j,f]
    v8f acc = {};
    if (col < B_SZ) {
      const float* cp = base + (((size_t)col * N_NODES + i) * N_NODES + j) * F_FEAT + half * 8;
      v4f c0 = *(const v4f*)(cp);
      v4f c1 = *(const v4f*)(cp + 4);
      acc[0] = c0.x; acc[1] = c0.y; acc[2] = c0.z; acc[3] = c0.w;
      acc[4] = c1.x; acc[5] = c1.y; acc[6] = c1.z; acc[7] = c1.w;
    }
    acc = wmma_k16(aw, bagg, acc);

#pragma unroll
    for (int v = 0; v < 8; ++v)
      msum[v] += aij * fmaxf(acc[v], 0.f);
  }

  // deterministic cross-wave reduction
  __shared__ float red[8][32][8];
#pragma unroll
  for (int v = 0; v < 8; ++v) red[wave][lane][v] = msum[v];
  __syncthreads();
  if (wave == 0 && col < B_SZ) {
#pragma unroll
    for (int v = 0; v < 8; ++v) {
      float s = 0.f;
#pragma unroll
      for (int w = 0; w < 8; ++w) s += red[w][lane][v];
      msg_sum[((size_t)col * N_NODES + i) * F_FEAT + half * 8 + v] = s;
    }
  }
}

// ---- Phase D1: enc[b, i*16+f] = relu(U_nf[i]@x[b,i] + U_nm[i]@msg_sum[b,i])
__global__ void __launch_bounds__(256) phaseD1(const float* __restrict__ x,
                                               const float* __restrict__ u_nf,
                                               const float* __restrict__ u_nm,
                                               const float* __restrict__ msg_sum,
                                               float* __restrict__ enc) {
  const int lane = threadIdx.x & 31;
  const int wave = threadIdx.x >> 5;
  const int col  = lane & 15;
  const int half = lane >> 4;
  const int i = blockIdx.x * 8 + wave;

  v2f bx[4], bm[4];
#pragma unroll
  for (int c = 0; c < 4; ++c) {
    const int g = c * 4 + half * 2;
    v2f vx = {0.f, 0.f}, vm = {0.f, 0.f};
    if (col < B_SZ) {
      vx = *(const v2f*)(x + (size_t)col * N_NODES * F_FEAT + (size_t)i * F_FEAT + g);
      vm = *(const v2f*)(msg_sum + ((size_t)col * N_NODES + i) * F_FEAT + g);
    }
    bx[c] = vx; bm[c] = vm;
  }

  v2f a1[4], a2[4];
  load_w16(u_nf + (size_t)i * (F_FEAT * F_FEAT), col, half, a1);
  load_w16(u_nm + (size_t)i * (F_FEAT * F_FEAT), col, half, a2);

  v8f acc = {};
  acc = wmma_k16(a1, bx, acc);
  acc = wmma_k16(a2, bm, acc);

  if (col < B_SZ) {
    float* ep = enc + (size_t)col * FCIN + (size_t)i * F_FEAT + half * 8;
    v4f e0 = {fmaxf(acc[0], 0.f), fmaxf(acc[1], 0.f), fmaxf(acc[2], 0.f), fmaxf(acc[3], 0.f)};
    v4f e1 = {fmaxf(acc[4], 0.f), fmaxf(acc[5], 0.f), fmaxf(acc[6], 0.f), fmaxf(acc[7], 0.f)};
    *(v4f*)(ep)     = e0;
    *(v4f*)(ep + 4) = e1;
  }
}

// ---- Phase D2: out[b,o] = sigmoid(enc[b,:] . lin_w[o,:] + lin_b[o]) ------
__global__ void __launch_bounds__(256) phaseD2(const float* __restrict__ enc,
                                               const float* __restrict__ lin_w,
                                               const float* __restrict__ lin_b,
                                               float* __restrict__ out) {
  const int tid = blockIdx.x * blockDim.x + threadIdx.x;
  if (tid >= B_SZ * FCOUT) return;
  const int b = tid & (B_SZ - 1);
  const int o = tid >> 3;
  const float* ep = enc + (size_t)b * FCIN;
  const float* wp = lin_w + (size_t)o * FCIN;
  float acc = lin_b[o];
  for (int t = 0; t < FCIN; ++t) acc += ep[t] * wp[t];
  out[b * FCOUT + o] = 1.f / (1.f + expf(-acc));
}

// ---- launcher ------------------------------------------------------------
extern "C" void kernel_launch(void* const* d_in, const int* in_sizes, int n_in,
                              void* d_out, int out_size, void* d_ws, size_t ws_size,
                              hipStream_t stream) {
  const float* x     = (const float*)d_in[0];
  const float* Aadj  = (const float*)d_in[1];
  // d_in[2] = batch_size scalar (compile-time B_SZ)
  const float* w_nf  = (const float*)d_in[3];
  const float* w_nm  = (const float*)d_in[4];
  const float* u_nf  = (const float*)d_in[5];
  const float* u_nm  = (const float*)d_in[6];
  const float* lin_w = (const float*)d_in[7];
  const float* lin_b = (const float*)d_in[8];
  float* out = (float*)d_out;

  // workspace layout (all fully overwritten every call; no init needed)
  const size_t base_elems = (size_t)B_SZ * N_NODES * N_NODES * F_FEAT; // 33.5M floats, 128MB
  const size_t vec_elems  = (size_t)B_SZ * N_NODES * F_FEAT;           // 64K floats
  float* base     = (float*)d_ws;
  float* incoming = base + base_elems;
  float* msg_sum  = incoming + vec_elems;
  float* enc      = msg_sum + vec_elems;

  phaseA <<<dim3(N_NODES, N_NODES / 8), 256, 0, stream>>>(x, w_nf, base);
  phaseB <<<dim3(N_NODES, B_SZ),        256, 0, stream>>>(Aadj, base, incoming);
  phaseC <<<dim3(N_NODES),              256, 0, stream>>>(Aadj, w_nm, base, incoming, msg_sum);
  phaseD1<<<dim3(N_NODES / 8),          256, 0, stream>>>(x, u_nf, u_nm, msg_sum, enc);
  phaseD2<<<dim3(4),                    256, 0, stream>>>(enc, lin_w, lin_b, out);
}